// ControllerNet_85452669321958
// MI455X (gfx1250) — compile-verified
//
#include <hip/hip_runtime.h>

typedef float v2f __attribute__((ext_vector_type(2)));
typedef float v8f __attribute__((ext_vector_type(8)));

#define HIDDEN 2048
#define EMBED  512
#define NV     16
#define WAVES  16
#define BLOCK  (WAVES * 32)

// D = A(16x4 f32) * B(4x16 f32) + C(16x16 f32)
__device__ __forceinline__ v8f wmma_f32_k4(v2f a, v2f b, v8f c) {
  return __builtin_amdgcn_wmma_f32_16x16x4_f32(false, a, false, b, (short)0, c,
                                               false, false);
}

// Three fused GEMV tiles (the r/z/n gate row-blocks, HIDDEN rows apart) over
// one shared vector segment. A-fragment: lane L<16 = row (k0,k1), lane L+16 =
// row (k2,k3) -> direct per-lane b64 load at (row, kb + 2*hi); no shuffles.
// B replicated over N: lanes<16 load (x[k0],x[k1]) = rows K0/K1, lanes>=16
// load (x[k2],x[k3]) = rows K2/K3.
template <int KPER>
__device__ __forceinline__ void gemv16x3_wmma(const float* __restrict__ W,
                                              int stride, int jBase,
                                              const float* __restrict__ sX,
                                              int kStart,
                                              v8f& c0, v8f& c1, v8f& c2) {
  const int L   = threadIdx.x & 31;
  const int sub = L & 15;
  const int hi2 = (L >> 4) << 1;  // 0 or 2
  const float* r0 = W + (size_t)(jBase + sub) * (size_t)stride + hi2 + kStart;
  const float* r1 = r0 + (size_t)HIDDEN * (size_t)stride;
  const float* r2 = r1 + (size_t)HIDDEN * (size_t)stride;
  const float* xp = sX + hi2 + kStart;
#pragma unroll 4
  for (int kb = 0; kb < KPER; kb += 4) {
    const float2 b  = *(const float2*)(xp + kb);
    const float2 a0 = *(const float2*)(r0 + kb);
    const float2 a1 = *(const float2*)(r1 + kb);
    const float2 a2 = *(const float2*)(r2 + kb);
    v2f B;  B.x  = b.x;  B.y  = b.y;
    v2f A0; A0.x = a0.x; A0.y = a0.y;
    v2f A1; A1.x = a1.x; A1.y = a1.y;
    v2f A2; A2.x = a2.x; A2.y = a2.y;
    c0 = wmma_f32_k4(A0, B, c0);   // independent accumulators: no
    c1 = wmma_f32_k4(A1, B, c1);   // back-to-back WMMA RAW hazard
    c2 = wmma_f32_k4(A2, B, c2);
  }
}

// Single-tile variant for the 16-row softmax head.
template <int KPER>
__device__ __forceinline__ void gemv16_wmma(const float* __restrict__ W,
                                            int stride,
                                            const float* __restrict__ sX,
                                            int kStart, v8f& c) {
  const int L   = threadIdx.x & 31;
  const int sub = L & 15;
  const int hi2 = (L >> 4) << 1;
  const float* r0 = W + (size_t)sub * (size_t)stride + hi2 + kStart;
  const float* xp = sX + hi2 + kStart;
#pragma unroll 4
  for (int kb = 0; kb < KPER; kb += 4) {
    const float2 b = *(const float2*)(xp + kb);
    const float2 a = *(const float2*)(r0 + kb);
    v2f B; B.x = b.x; B.y = b.y;
    v2f A; A.x = a.x; A.y = a.y;
    c = wmma_f32_k4(A, B, c);
  }
}

// C layout: element (M,N) lives in vgpr (M%8), lane N + 16*(M/8); values are
// replicated over N, so lane m (and lane 16+m) publish rows m / m+8.
__device__ __forceinline__ void reduce16(v8f c, float* acc) {
  const int L   = threadIdx.x & 31;
  const int sub = L & 15;
  const int hi  = L >> 4;
#pragma unroll
  for (int m = 0; m < 8; ++m) {
    if (sub == m) atomicAdd(&acc[m + 8 * hi], c[m]);  // ds_add_f32
  }
}

// grid (HIDDEN/16, 2): blockIdx.y = 0 -> input-hidden matrix, 1 -> hidden-
// hidden matrix. Each block streams three gate row-blocks of one matrix and
// writes bias-added pre-activations to pre[(mat*3+g)*HIDDEN + j] (single
// writer per element; no init required).
template <int K_IH, int K_HH>
__global__ void __launch_bounds__(BLOCK)
gru_matvec_kernel(const float* __restrict__ W_ih, const float* __restrict__ W_hh,
                  const float* __restrict__ b_ih, const float* __restrict__ b_hh,
                  const float* __restrict__ xsrc, const int* __restrict__ xidx,
                  const float* __restrict__ hprev, float* __restrict__ pre) {
  __shared__ float sX[HIDDEN];
  __shared__ float sAcc[3][16];

  const int tid   = threadIdx.x;
  const int jBase = blockIdx.x * 16;
  const int mat   = blockIdx.y;

  if (mat == 0) {
    const float* xv = xidx ? (xsrc + (size_t)xidx[0] * (size_t)K_IH) : xsrc;
    for (int i = tid; i < K_IH; i += BLOCK) sX[i] = xv[i];
  } else {
    for (int i = tid; i < K_HH; i += BLOCK) sX[i] = hprev[i];
  }
  if (tid < 48) ((float*)sAcc)[tid] = 0.0f;
  __syncthreads();

  // Scalarize wave id: k-loop bounds in SGPRs -> scalar loop, EXEC all-ones.
  const int wave = __builtin_amdgcn_readfirstlane(tid >> 5);

  if (mat == 0) {
    constexpr int kPer = K_IH / WAVES;
    v8f c0 = {}, c1 = {}, c2 = {};
    gemv16x3_wmma<kPer>(W_ih, K_IH, jBase, sX, wave * kPer, c0, c1, c2);
    reduce16(c0, sAcc[0]);
    reduce16(c1, sAcc[1]);
    reduce16(c2, sAcc[2]);
  } else {
    constexpr int kPer = K_HH / WAVES;
    v8f c0 = {}, c1 = {}, c2 = {};
    gemv16x3_wmma<kPer>(W_hh, K_HH, jBase, sX, wave * kPer, c0, c1, c2);
    reduce16(c0, sAcc[0]);
    reduce16(c1, sAcc[1]);
    reduce16(c2, sAcc[2]);
  }
  __syncthreads();

  if (tid < 48) {
    const int g = tid >> 4;        // gate 0..2
    const int m = tid & 15;
    const int j = jBase + m;
    const float* bias = (mat == 0) ? b_ih : b_hh;
    pre[(mat * 3 + g) * HIDDEN + j] = sAcc[g][m] + bias[g * HIDDEN + j];
  }
}

__global__ void __launch_bounds__(256)
gru_gate_kernel(const float* __restrict__ pre, const float* __restrict__ hprev,
                float* __restrict__ hnext, float* __restrict__ hcopy) {
  const int j = blockIdx.x * 256 + threadIdx.x;
  if (j < HIDDEN) {
    const float ir  = pre[0 * HIDDEN + j];
    const float iz  = pre[1 * HIDDEN + j];
    const float inn = pre[2 * HIDDEN + j];
    const float hr  = pre[3 * HIDDEN + j];
    const float hz  = pre[4 * HIDDEN + j];
    const float hn  = pre[5 * HIDDEN + j];
    const float r = 1.0f / (1.0f + __expf(-(ir + hr)));
    const float z = 1.0f / (1.0f + __expf(-(iz + hz)));
    const float n = tanhf(inn + r * hn);
    const float h = (1.0f - z) * n + z * hprev[j];
    hnext[j] = h;   // workspace: feeds next layer / head
    hcopy[j] = h;   // d_out h_out slice
  }
}

__global__ void __launch_bounds__(BLOCK)
head_kernel(const float* __restrict__ w_head, const float* __restrict__ b_head,
            const float* __restrict__ h1, float* __restrict__ probs) {
  __shared__ float sX[HIDDEN];
  __shared__ float sAcc[16];
  const int tid = threadIdx.x;
  for (int i = tid; i < HIDDEN; i += BLOCK) sX[i] = h1[i];
  if (tid < 16) sAcc[tid] = 0.0f;
  __syncthreads();

  const int wave = __builtin_amdgcn_readfirstlane(tid >> 5);
  constexpr int kPer = HIDDEN / WAVES;
  v8f c = {};
  gemv16_wmma<kPer>(w_head, HIDDEN, sX, wave * kPer, c);
  reduce16(c, sAcc);
  __syncthreads();

  if (tid < 16) {
    const float v = sAcc[tid] + b_head[tid];
    float m = v;
    for (int off = 8; off > 0; off >>= 1) m = fmaxf(m, __shfl_xor(m, off, 32));
    const float e = __expf(v - m);
    float s = e;
    for (int off = 8; off > 0; off >>= 1) s += __shfl_xor(s, off, 32);
    probs[tid] = e / s;
  }
}

extern "C" void kernel_launch(void* const* d_in, const int* in_sizes, int n_in,
                              void* d_out, int out_size, void* d_ws,
                              size_t ws_size, hipStream_t stream) {
  const int*   x      = (const int*)  d_in[0];
  const float* h_in   = (const float*)d_in[1];   // [2,1,2048]
  const float* emb    = (const float*)d_in[2];   // [64,512]
  const float* w_ih0  = (const float*)d_in[3];   // [6144,512]
  const float* w_hh0  = (const float*)d_in[4];   // [6144,2048]
  const float* b_ih0  = (const float*)d_in[5];
  const float* b_hh0  = (const float*)d_in[6];
  const float* w_ih1  = (const float*)d_in[7];   // [6144,2048]
  const float* w_hh1  = (const float*)d_in[8];   // [6144,2048]
  const float* b_ih1  = (const float*)d_in[9];
  const float* b_hh1  = (const float*)d_in[10];
  const float* w_head = (const float*)d_in[11];  // [16,2048]
  const float* b_head = (const float*)d_in[12];

  float* out = (float*)d_out;  // [0..15] probs, [16..2063] h0, [2064..4111] h1
  float* h0  = (float*)d_ws;
  float* h1  = h0 + HIDDEN;
  float* pre = h1 + HIDDEN;    // [6][HIDDEN] pre-activations (reused per layer)

  const dim3 gridMV(HIDDEN / 16, 2);

  gru_matvec_kernel<EMBED, HIDDEN><<<gridMV, BLOCK, 0, stream>>>(
      w_ih0, w_hh0, b_ih0, b_hh0, emb, x, h_in, pre);
  gru_gate_kernel<<<HIDDEN / 256, 256, 0, stream>>>(pre, h_in, h0, out + NV);

  gru_matvec_kernel<HIDDEN, HIDDEN><<<gridMV, BLOCK, 0, stream>>>(
      w_ih1, w_hh1, b_ih1, b_hh1, h0, nullptr, h_in + HIDDEN, pre);
  gru_gate_kernel<<<HIDDEN / 256, 256, 0, stream>>>(pre, h_in + HIDDEN, h1,
                                                    out + NV + HIDDEN);

  head_kernel<<<1, BLOCK, 0, stream>>>(w_head, b_head, h1, out);
}